// GATpaper_87548613362020
// MI455X (gfx1250) — compile-verified
//
#include <hip/hip_runtime.h>

typedef __attribute__((ext_vector_type(2)))  float    v2f;
typedef __attribute__((ext_vector_type(8)))  float    v8f;

#define FDIM 6
#define PAD  8   // padded row stride for xl/xr (32B -> b128 gathers)

__device__ __forceinline__ float leakyf(float v, float s) {
    return v >= 0.0f ? v : s * v;
}

// Monotone float->uint mapping for atomicMax-based segment max.
__device__ __forceinline__ unsigned ord_enc(float f) {
    unsigned u = __float_as_uint(f);
    return (u & 0x80000000u) ? ~u : (u | 0x80000000u);
}
__device__ __forceinline__ float ord_dec(unsigned u) {
    return (u & 0x80000000u) ? __uint_as_float(u & 0x7FFFFFFFu)
                             : __uint_as_float(~u);
}

// ---------------------------------------------------------------------------
// 16-node x 16-col x K<=8 f32 tile matmul via V_WMMA_F32_16X16X4_F32 (x2).
// A(m,k) supplied by Xv, B(k,n) supplied by Bv.  C layout: lane holds column
// n = lane&15; VGPR r holds row m = r + 8*(lane>>4).
// ---------------------------------------------------------------------------
template <typename FA, typename FB>
__device__ __forceinline__ v8f wmma_tile_k8(FA Xv, FB Bv, int half) {
    v8f c = {};
    v2f a0, a1, b0, b1;
    a0.x = Xv(2 * half);     a0.y = Xv(2 * half + 1);
    a1.x = Xv(4 + 2 * half); a1.y = Xv(5 + 2 * half);
    b0.x = Bv(2 * half);     b0.y = Bv(2 * half + 1);
    b1.x = Bv(4 + 2 * half); b1.y = Bv(5 + 2 * half);
    c = __builtin_amdgcn_wmma_f32_16x16x4_f32(false, a0, false, b0, (short)0, c, false, false);
    c = __builtin_amdgcn_wmma_f32_16x16x4_f32(false, a1, false, b1, (short)0, c, false, false);
    return c;
}

// ---------------------------------------------------------------------------
// xl = x2 @ Wl + bl ; xr = x2 @ Wr + br   (Wl in cols 0-5, Wr in cols 8-13)
// xl/xr stored with row stride PAD=8; padding cols carry WMMA's exact zeros.
// ---------------------------------------------------------------------------
__global__ void k_transform(const float* __restrict__ x2,
                            const float* __restrict__ Wl, const float* __restrict__ bl,
                            const float* __restrict__ Wr, const float* __restrict__ br,
                            float* __restrict__ xl, float* __restrict__ xr, int nN) {
    int lane = threadIdx.x & 31;
    int wave = (blockIdx.x * blockDim.x + threadIdx.x) >> 5;
    int base = wave * 16;
    if (base >= nN) return;
    int half = lane >> 4;
    int n = lane & 15;
    int m = base + n;
    bool mval = (m < nN);
    auto Xv = [&](int k) -> float { return (k < FDIM && mval) ? x2[m * FDIM + k] : 0.0f; };
    auto Bv = [&](int k) -> float {
        if (k >= FDIM) return 0.0f;
        if (n < FDIM) return Wl[k * FDIM + n];
        if (n >= 8 && n < 8 + FDIM) return Wr[k * FDIM + (n - 8)];
        return 0.0f;
    };
    v8f c = wmma_tile_k8(Xv, Bv, half);
    // per-lane bias, loaded once
    float bias = (n < 8) ? ((n < FDIM) ? bl[n] : 0.0f)
                         : ((n - 8 < FDIM) ? br[n - 8] : 0.0f);
    float* dst = (n < 8) ? xl : xr;
    int    col = n & 7;
    #pragma unroll
    for (int r = 0; r < 8; ++r) {
        int row = base + r + 8 * half;
        if (row < nN) dst[row * PAD + col] = c[r] + bias;
    }
}

// ---------------------------------------------------------------------------
// t = leaky(x1n @ W + b, 0.01)          (stride-6 in / stride-6 out)
// ---------------------------------------------------------------------------
__global__ void k_ff1(const float* __restrict__ xin,
                      const float* __restrict__ W, const float* __restrict__ b,
                      float* __restrict__ tout, int nN) {
    int lane = threadIdx.x & 31;
    int wave = (blockIdx.x * blockDim.x + threadIdx.x) >> 5;
    int base = wave * 16;
    if (base >= nN) return;
    int half = lane >> 4;
    int n = lane & 15;
    int m = base + n;
    bool mval = (m < nN);
    auto Xv = [&](int k) -> float { return (k < FDIM && mval) ? xin[m * FDIM + k] : 0.0f; };
    auto Bv = [&](int k) -> float { return (k < FDIM && n < FDIM) ? W[k * FDIM + n] : 0.0f; };
    v8f c = wmma_tile_k8(Xv, Bv, half);
    if (n >= FDIM) return;
    float bias = b[n];
    #pragma unroll
    for (int r = 0; r < 8; ++r) {
        int row = base + r + 8 * half;
        if (row < nN) tout[row * FDIM + n] = leakyf(c[r] + bias, 0.01f);
    }
}

// ---------------------------------------------------------------------------
// h2 = t @ W + b + r0 [+ r1 + r2 + r3]
// ---------------------------------------------------------------------------
__global__ void k_ff2(const float* __restrict__ tin,
                      const float* __restrict__ W, const float* __restrict__ b,
                      const float* __restrict__ r0, const float* __restrict__ r1,
                      const float* __restrict__ r2, const float* __restrict__ r3,
                      float* __restrict__ h2, int nN) {
    int lane = threadIdx.x & 31;
    int wave = (blockIdx.x * blockDim.x + threadIdx.x) >> 5;
    int base = wave * 16;
    if (base >= nN) return;
    int half = lane >> 4;
    int n = lane & 15;
    int m = base + n;
    bool mval = (m < nN);
    auto Xv = [&](int k) -> float { return (k < FDIM && mval) ? tin[m * FDIM + k] : 0.0f; };
    auto Bv = [&](int k) -> float { return (k < FDIM && n < FDIM) ? W[k * FDIM + n] : 0.0f; };
    v8f c = wmma_tile_k8(Xv, Bv, half);
    if (n >= FDIM) return;
    float bias = b[n];
    #pragma unroll
    for (int r = 0; r < 8; ++r) {
        int row = base + r + 8 * half;
        if (row < nN) {
            int i = row * FDIM + n;
            float v = c[r] + bias + r0[i];
            if (r1) v += r1[i];
            if (r2) v += r2[i];
            if (r3) v += r3[i];
            h2[i] = v;
        }
    }
}

// ---------------------------------------------------------------------------
// Edge pass A: logit per edge + segment max over dst.  b128 gathers.
// ---------------------------------------------------------------------------
__global__ void k_edge_logit(const float* __restrict__ xl, const float* __restrict__ xr,
                             const float* __restrict__ ea, const int* __restrict__ eidx,
                             const float* __restrict__ We, const float* __restrict__ att,
                             float* __restrict__ elog, unsigned* __restrict__ lmax, int E) {
    int e = blockIdx.x * blockDim.x + threadIdx.x;
    if (e >= E) return;
    int s = eidx[e];
    int d = eidx[E + e];
    float2 ev = ((const float2*)ea)[e];
    const float4* xl4 = (const float4*)xl;
    const float4* xr4 = (const float4*)xr;
    float4 sa = xl4[s * 2], sb = xl4[s * 2 + 1];
    float4 da = xr4[d * 2], db = xr4[d * 2 + 1];
    float sx[FDIM] = { sa.x, sa.y, sa.z, sa.w, sb.x, sb.y };
    float dx[FDIM] = { da.x, da.y, da.z, da.w, db.x, db.y };
    float lg = 0.0f;
    #pragma unroll
    for (int f = 0; f < FDIM; ++f) {
        float ef = ev.x * We[f] + ev.y * We[FDIM + f];
        float m = leakyf(sx[f] + dx[f] + ef, 0.2f);
        lg += m * att[f];
    }
    elog[e] = lg;
    atomicMax(&lmax[d], ord_enc(lg));
}

// ---------------------------------------------------------------------------
// Edge pass B: a = exp(logit - lmax[dst]); denom[dst] += a  (a overwrites elog)
// ---------------------------------------------------------------------------
__global__ void k_edge_expsum(float* __restrict__ elog, const unsigned* __restrict__ lmax,
                              float* __restrict__ denom, const int* __restrict__ eidx, int E) {
    int e = blockIdx.x * blockDim.x + threadIdx.x;
    if (e >= E) return;
    int d = eidx[E + e];
    float a = __expf(elog[e] - ord_dec(lmax[d]));
    elog[e] = a;
    atomicAdd(&denom[d], a);
}

// ---------------------------------------------------------------------------
// Edge pass C: hacc[dst] += xl[src] * (a / denom[dst])
// ---------------------------------------------------------------------------
__global__ void k_edge_accum(const float* __restrict__ elog, const float* __restrict__ denom,
                             const float* __restrict__ xl, const int* __restrict__ eidx,
                             float* __restrict__ hacc, int E) {
    int e = blockIdx.x * blockDim.x + threadIdx.x;
    if (e >= E) return;
    int s = eidx[e];
    int d = eidx[E + e];
    float w = elog[e] / denom[d];
    const float4* xl4 = (const float4*)xl;
    float4 sa = xl4[s * 2], sb = xl4[s * 2 + 1];
    float sx[FDIM] = { sa.x, sa.y, sa.z, sa.w, sb.x, sb.y };
    #pragma unroll
    for (int f = 0; f < FDIM; ++f) {
        atomicAdd(&hacc[d * FDIM + f], sx[f] * w);
    }
}

// ---------------------------------------------------------------------------
// h = leaky(hacc + bias_g, 0.01) + residuals
// ---------------------------------------------------------------------------
__global__ void k_post_gat(const float* __restrict__ hacc, const float* __restrict__ bias,
                           const float* __restrict__ r0, const float* __restrict__ r1,
                           const float* __restrict__ r2, float* __restrict__ hbuf, int total) {
    int i = blockIdx.x * blockDim.x + threadIdx.x;
    if (i >= total) return;
    int col = i % FDIM;
    float v = leakyf(hacc[i] + bias[col], 0.01f);
    v += r0[i];
    if (r1) v += r1[i];
    if (r2) v += r2[i];
    hbuf[i] = v;
}

// ---------------------------------------------------------------------------
// Grid reduction of sum / sumsq into stats[0], stats[1].
// ---------------------------------------------------------------------------
__global__ void k_reduce_stats(const float* __restrict__ src, int n, float* __restrict__ stats) {
    float s = 0.0f, ss = 0.0f;
    for (int i = blockIdx.x * blockDim.x + threadIdx.x; i < n; i += gridDim.x * blockDim.x) {
        float v = src[i];
        s += v; ss += v * v;
    }
    #pragma unroll
    for (int off = 16; off > 0; off >>= 1) {
        s  += __shfl_xor(s,  off, 32);
        ss += __shfl_xor(ss, off, 32);
    }
    __shared__ float sm[16];
    int wid = threadIdx.x >> 5, lane = threadIdx.x & 31;
    int nw = blockDim.x >> 5;
    if (lane == 0) { sm[wid] = s; sm[8 + wid] = ss; }
    __syncthreads();
    if (wid == 0) {
        s  = (lane < nw) ? sm[lane]     : 0.0f;
        ss = (lane < nw) ? sm[8 + lane] : 0.0f;
        #pragma unroll
        for (int off = 4; off > 0; off >>= 1) {
            s  += __shfl_xor(s,  off, 32);
            ss += __shfl_xor(ss, off, 32);
        }
        if (lane == 0) { atomicAdd(&stats[0], s); atomicAdd(&stats[1], ss); }
    }
}

__global__ void k_finalize_stats(float* __restrict__ stats, int total) {
    if (threadIdx.x == 0 && blockIdx.x == 0) {
        float s = stats[0], ss = stats[1];
        float cnt = (float)total;
        float mean = s / cnt;
        float var = (ss - s * s / cnt) / (cnt - 1.0f);
        stats[2] = mean;
        stats[3] = rsqrtf(var);
    }
}

__global__ void k_norm_apply(float* __restrict__ dst, const float* __restrict__ src,
                             const float* __restrict__ stats, int total) {
    int i = blockIdx.x * blockDim.x + threadIdx.x;
    if (i >= total) return;
    dst[i] = (src[i] - stats[2]) * stats[3];
}

__global__ void k_zero(float* __restrict__ p, int n) {
    int i = blockIdx.x * blockDim.x + threadIdx.x;
    if (i < n) p[i] = 0.0f;
}

// ---------------------------------------------------------------------------
// Reghead collapse: c1 = Wresh@W1, c2 = Wresh@W2, d1 = bresh.W1+b1, d2 = ...
// head[0..5]=c1, head[6..11]=c2, head[12]=d1, head[13]=d2
// ---------------------------------------------------------------------------
__global__ void k_head_precompute(const float* __restrict__ Wresh, const float* __restrict__ bresh,
                                  const float* __restrict__ W1, const float* __restrict__ b1,
                                  const float* __restrict__ W2, const float* __restrict__ b2,
                                  float* __restrict__ head, int RG) {
    __shared__ float acc[14];
    if (threadIdx.x < 14) acc[threadIdx.x] = 0.0f;
    __syncthreads();
    float p[14];
    #pragma unroll
    for (int q = 0; q < 14; ++q) p[q] = 0.0f;
    for (int k = threadIdx.x; k < RG; k += blockDim.x) {
        float w1 = W1[k], w2 = W2[k], bb = bresh[k];
        #pragma unroll
        for (int j = 0; j < FDIM; ++j) {
            float wr = Wresh[j * RG + k];
            p[j]        += wr * w1;
            p[FDIM + j] += wr * w2;
        }
        p[12] += bb * w1;
        p[13] += bb * w2;
    }
    #pragma unroll
    for (int q = 0; q < 14; ++q) atomicAdd(&acc[q], p[q]);
    __syncthreads();
    if (threadIdx.x < 14) {
        float extra = (threadIdx.x == 12) ? b1[0] : (threadIdx.x == 13) ? b2[0] : 0.0f;
        head[threadIdx.x] = acc[threadIdx.x] + extra;
    }
}

__global__ void k_reghead(const float* __restrict__ h2, const float* __restrict__ head,
                          float* __restrict__ out, int nN) {
    int n = blockIdx.x * blockDim.x + threadIdx.x;
    if (n >= nN) return;
    const float* h = h2 + n * FDIM;
    float vr = head[12], vi = head[13];
    #pragma unroll
    for (int f = 0; f < FDIM; ++f) {
        vr += h[f] * head[f];
        vi += h[f] * head[FDIM + f];
    }
    out[n * 2]     = vr;
    out[n * 2 + 1] = vi;
}

// ---------------------------------------------------------------------------
extern "C" void kernel_launch(void* const* d_in, const int* in_sizes, int n_in,
                              void* d_out, int out_size, void* d_ws, size_t ws_size,
                              hipStream_t stream) {
    const float* x         = (const float*)d_in[0];
    const float* edge_attr = (const float*)d_in[1];
    const float* Wl        = (const float*)d_in[2];
    const float* bl        = (const float*)d_in[3];
    const float* Wr        = (const float*)d_in[4];
    const float* br        = (const float*)d_in[5];
    const float* We        = (const float*)d_in[6];
    const float* att       = (const float*)d_in[7];
    const float* bias_g    = (const float*)d_in[8];
    const float* ff1W      = (const float*)d_in[9];
    const float* ff1b      = (const float*)d_in[10];
    const float* ff2W      = (const float*)d_in[11];
    const float* ff2b      = (const float*)d_in[12];
    const float* Wresh     = (const float*)d_in[13];
    const float* bresh     = (const float*)d_in[14];
    const float* W1        = (const float*)d_in[15];
    const float* b1        = (const float*)d_in[16];
    const float* W2        = (const float*)d_in[17];
    const float* b2        = (const float*)d_in[18];
    const int*   eidx      = (const int*)  d_in[19];

    const int N_  = in_sizes[0] / FDIM;
    const int E_  = in_sizes[19] / 2;
    const int RG_ = in_sizes[14];
    const int NF  = N_ * FDIM;
    const int NP  = N_ * PAD;
    float* out = (float*)d_out;

    // ---- workspace layout (floats) ----
    float* p = (float*)d_ws;
    float* xl    = p; p += NP;   // padded stride 8
    float* xr    = p; p += NP;   // padded stride 8
    float* hbuf  = p; p += NF;
    float* x1nA  = p; p += NF;   // layer-0 norm output (x1 / x3)
    float* x1nB  = p; p += NF;   // layer-1 norm output
    float* x2buf = p; p += NF;   // norm(h2) after layer 0
    float* h2buf = p; p += NF;
    float* tbuf  = p; p += NF;
    float* hacc  = p; p += NF;   // --- start of zero-region ---
    unsigned* lmax = (unsigned*)p; p += N_;
    float* denom = p; p += N_;
    float* stats = p; p += 8;    // --- end of zero-region ---
    float* head  = p; p += 16;
    float* elog  = p; p += E_;
    const int zcount = NF + N_ + N_ + 8;

    const int B = 256;
    dim3 blk(B);
    dim3 gZero((zcount + B - 1) / B);
    dim3 gEdge((E_ + B - 1) / B);
    dim3 gElem((NF + B - 1) / B);
    dim3 gNode((N_ + B - 1) / B);
    dim3 gWmma((N_ + 63) / 64);   // 128 threads = 4 waves, 16 nodes/wave
    dim3 blkW(128);
    dim3 gRed(256);

    k_head_precompute<<<1, 256, 0, stream>>>(Wresh, bresh, W1, b1, W2, b2, head, RG_);

    // ======================= Layer 0 (x2 = x) =======================
    k_zero<<<gZero, blk, 0, stream>>>(hacc, zcount);
    k_transform<<<gWmma, blkW, 0, stream>>>(x, Wl, bl, Wr, br, xl, xr, N_);
    k_edge_logit<<<gEdge, blk, 0, stream>>>(xl, xr, edge_attr, eidx, We, att, elog, lmax, E_);
    k_edge_expsum<<<gEdge, blk, 0, stream>>>(elog, lmax, denom, eidx, E_);
    k_edge_accum<<<gEdge, blk, 0, stream>>>(elog, denom, xl, eidx, hacc, E_);
    k_post_gat<<<gElem, blk, 0, stream>>>(hacc, bias_g, x, nullptr, nullptr, hbuf, NF);
    k_reduce_stats<<<gRed, blk, 0, stream>>>(hbuf, NF, stats);
    k_finalize_stats<<<1, 32, 0, stream>>>(stats, NF);
    k_norm_apply<<<gElem, blk, 0, stream>>>(x1nA, hbuf, stats, NF);
    k_ff1<<<gWmma, blkW, 0, stream>>>(x1nA, ff1W, ff1b, tbuf, N_);
    k_ff2<<<gWmma, blkW, 0, stream>>>(tbuf, ff2W, ff2b, x1nA, x, nullptr, nullptr, h2buf, N_);
    k_zero<<<1, 32, 0, stream>>>(stats, 8);
    k_reduce_stats<<<gRed, blk, 0, stream>>>(h2buf, NF, stats);
    k_finalize_stats<<<1, 32, 0, stream>>>(stats, NF);
    k_norm_apply<<<gElem, blk, 0, stream>>>(x2buf, h2buf, stats, NF);

    // ======================= Layer 1 (x2 = x2buf) =======================
    const float* Wl1 = Wl + 36;  const float* bl1 = bl + 6;
    const float* Wr1 = Wr + 36;  const float* br1 = br + 6;
    const float* We1 = We + 12;  const float* att1 = att + 6;
    const float* bg1 = bias_g + 6;
    const float* f1W = ff1W + 36; const float* f1b = ff1b + 6;
    const float* f2W = ff2W + 36; const float* f2b = ff2b + 6;

    k_zero<<<gZero, blk, 0, stream>>>(hacc, zcount);
    k_transform<<<gWmma, blkW, 0, stream>>>(x2buf, Wl1, bl1, Wr1, br1, xl, xr, N_);
    k_edge_logit<<<gEdge, blk, 0, stream>>>(xl, xr, edge_attr, eidx, We1, att1, elog, lmax, E_);
    k_edge_expsum<<<gEdge, blk, 0, stream>>>(elog, lmax, denom, eidx, E_);
    k_edge_accum<<<gEdge, blk, 0, stream>>>(elog, denom, xl, eidx, hacc, E_);
    // h = leaky(gat) + x0 + x1 + x2
    k_post_gat<<<gElem, blk, 0, stream>>>(hacc, bg1, x, x1nA, x2buf, hbuf, NF);
    k_reduce_stats<<<gRed, blk, 0, stream>>>(hbuf, NF, stats);
    k_finalize_stats<<<1, 32, 0, stream>>>(stats, NF);
    k_norm_apply<<<gElem, blk, 0, stream>>>(x1nB, hbuf, stats, NF);
    k_ff1<<<gWmma, blkW, 0, stream>>>(x1nB, f1W, f1b, tbuf, N_);
    // h2 = ff2(t) + x1n + x2 + x3 + x0
    k_ff2<<<gWmma, blkW, 0, stream>>>(tbuf, f2W, f2b, x1nB, x2buf, x1nA, x, h2buf, N_);

    // ======================= Reghead =======================
    k_reghead<<<gNode, blk, 0, stream>>>(h2buf, head, out, N_);
}